// I2P_Fusion_44367012168417
// MI455X (gfx1250) — compile-verified
//
#include <hip/hip_runtime.h>
#include <math.h>

typedef __attribute__((ext_vector_type(2))) float v2f;
typedef __attribute__((ext_vector_type(8))) float v8f;

// ---------------------------------------------------------------------------
// Generic f32 WMMA GEMM:  Y[b] = act( A[b?] (OxI) * X[b] (IxM) + bias )
// One wave (32 lanes) per 16x16 output tile; K stepped by 4 via
// V_WMMA_F32_16X16X4_F32 (exact f32 math on the matrix pipe).
// Requires O%16==0, M%16==0, I%4==0 (true for every GEMM in this workload),
// so EXEC is all-1s as WMMA demands.
// ---------------------------------------------------------------------------
__global__ __launch_bounds__(32)
void gemm_wmma_f32(const float* __restrict__ A, long aStride,
                   const float* __restrict__ X,
                   const float* __restrict__ bias,
                   float* __restrict__ Y,
                   int O, int I, int M, int relu)
{
    const int lane  = threadIdx.x & 31;
    const int mTile = blockIdx.x << 4;
    const int oTile = blockIdx.y << 4;
    const int b     = blockIdx.z;

    const float* Ab = A + (long)b * aStride;        // aStride==0 -> shared weights
    const float* Xb = X + (long)b * I * M;
    float*       Yb = Y + (long)b * O * M;

    const int half = lane >> 4;     // 0: K pair {0,1}; 1: K pair {2,3}
    const int l    = lane & 15;
    const long arowBase = (long)(oTile + l) * I + 2 * half;  // A row per lane
    const int  bcol     = mTile + l;                          // B col per lane

    v8f c = {0.f, 0.f, 0.f, 0.f, 0.f, 0.f, 0.f, 0.f};
    for (int k0 = 0; k0 < I; k0 += 4) {
        v2f a, bf;
        a.x  = Ab[arowBase + k0];
        a.y  = Ab[arowBase + k0 + 1];
        const int kb = k0 + 2 * half;
        bf.x = Xb[(long)kb * M + bcol];
        bf.y = Xb[(long)(kb + 1) * M + bcol];
        c = __builtin_amdgcn_wmma_f32_16x16x4_f32(false, a, false, bf,
                                                  (short)0, c, false, false);
    }
#pragma unroll
    for (int v = 0; v < 8; ++v) {
        const int orow = oTile + half * 8 + v;
        float val = c[v];
        if (bias) val += bias[orow];
        if (relu) val = fmaxf(val, 0.f);
        Yb[(long)orow * M + bcol] = val;
    }
}

// Copy a (B,C,srcM) block into channel slot [dstOff, dstOff+C) of (B,dstC,M).
// srcM==1 broadcasts along M (used for global_feature).
__global__ void copy_into(float* __restrict__ dst, int dstC, int dstOff,
                          const float* __restrict__ src, int C, int M, int srcM,
                          int B)
{
    long i = (long)blockIdx.x * blockDim.x + threadIdx.x;
    long total = (long)B * C * M;
    if (i >= total) return;
    int m = (int)(i % M);
    long t = i / M;
    int c = (int)(t % C);
    int b = (int)(t / C);
    float v = src[((long)b * C + c) * srcM + (srcM == 1 ? 0 : m)];
    dst[((long)b * dstC + dstOff + c) * M + m] = v;
}

// Softmax along channel dim of (B,C,M); one thread per (b,m) column.
__global__ void softmax_ch(float* __restrict__ X, int C, int M, int B)
{
    int i = blockIdx.x * blockDim.x + threadIdx.x;
    if (i >= B * M) return;
    int b = i / M, m = i % M;
    float* col = X + (long)b * C * M + m;
    float mx = -3.402823466e+38f;
    for (int c = 0; c < C; ++c) mx = fmaxf(mx, col[(long)c * M]);
    float s = 0.f;
    for (int c = 0; c < C; ++c) {
        float e = __expf(col[(long)c * M] - mx);
        col[(long)c * M] = e;
        s += e;
    }
    float inv = 1.f / s;
    for (int c = 0; c < C; ++c) col[(long)c * M] *= inv;
}

// Top-3 nearest key nodes per query point + interpolation weights
// w_k = 1 - d_k / (d0+d1+d2).  qpos: (B,3,Nq)  kpos: (B,3,Mk)
__global__ void topk3_weights(const float* __restrict__ qpos,
                              const float* __restrict__ kpos,
                              int Nq, int Mk,
                              int* __restrict__ idx, float* __restrict__ w, int B)
{
    int i = blockIdx.x * blockDim.x + threadIdx.x;
    if (i >= B * Nq) return;
    int b = i / Nq, n = i % Nq;
    const float* q = qpos + (long)b * 3 * Nq;
    float qx = q[n], qy = q[Nq + n], qz = q[2 * Nq + n];
    const float* kp = kpos + (long)b * 3 * Mk;
    float d0 = 3.4e38f, d1 = 3.4e38f, d2 = 3.4e38f;
    int   i0 = 0,       i1 = 0,       i2 = 0;
    for (int m = 0; m < Mk; ++m) {
        float dx = qx - kp[m], dy = qy - kp[Mk + m], dz = qz - kp[2 * Mk + m];
        float d = sqrtf(dx * dx + dy * dy + dz * dz);
        if (d < d0)      { d2 = d1; i2 = i1; d1 = d0; i1 = i0; d0 = d; i0 = m; }
        else if (d < d1) { d2 = d1; i2 = i1; d1 = d; i1 = m; }
        else if (d < d2) { d2 = d;  i2 = m; }
    }
    float s = d0 + d1 + d2;
    float inv = 1.f / s;
    long o = (long)i * 3;
    idx[o + 0] = i0; idx[o + 1] = i1; idx[o + 2] = i2;
    w[o + 0] = 1.f - d0 * inv;
    w[o + 1] = 1.f - d1 * inv;
    w[o + 2] = 1.f - d2 * inv;
}

// Weights only (indices precomputed): used for pc -> node_a.
__global__ void weights_from_idx(const float* __restrict__ qpos,
                                 const float* __restrict__ kpos,
                                 const int* __restrict__ idx,
                                 int Nq, int Mk, float* __restrict__ w, int B)
{
    int i = blockIdx.x * blockDim.x + threadIdx.x;
    if (i >= B * Nq) return;
    int b = i / Nq, n = i % Nq;
    const float* q = qpos + (long)b * 3 * Nq;
    float qx = q[n], qy = q[Nq + n], qz = q[2 * Nq + n];
    const float* kp = kpos + (long)b * 3 * Mk;
    long o = (long)i * 3;
    float d[3];
#pragma unroll
    for (int k = 0; k < 3; ++k) {
        int m = idx[o + k];
        float dx = qx - kp[m], dy = qy - kp[Mk + m], dz = qz - kp[2 * Mk + m];
        d[k] = sqrtf(dx * dx + dy * dy + dz * dz);
    }
    float inv = 1.f / (d[0] + d[1] + d[2]);
#pragma unroll
    for (int k = 0; k < 3; ++k) w[o + k] = 1.f - d[k] * inv;
}

// out[b,c,n] = sum_k w[b,n,k] * feat[b,c,idx[b,n,k]]
// Writes coalesced along n; feat rows (Mk<=256 floats) stay cache-resident.
__global__ void interp3(float* __restrict__ out, const float* __restrict__ feat,
                        const int* __restrict__ idx, const float* __restrict__ w,
                        int C, int Nq, int Mk, int B)
{
    int n = blockIdx.x * blockDim.x + threadIdx.x;
    int c = blockIdx.y;
    int b = blockIdx.z;
    if (n >= Nq) return;
    long o = ((long)b * Nq + n) * 3;
    const float* f = feat + ((long)b * C + c) * Mk;
    float acc = w[o] * f[idx[o]] + w[o + 1] * f[idx[o + 1]] + w[o + 2] * f[idx[o + 2]];
    out[((long)b * C + c) * Nq + n] = acc;
}

// ---------------------------------------------------------------------------

extern "C" void kernel_launch(void* const* d_in, const int* in_sizes, int n_in,
                              void* d_out, int out_size, void* d_ws, size_t ws_size,
                              hipStream_t stream)
{
    (void)in_sizes; (void)n_in; (void)out_size; (void)ws_size;
    const int B = 4, N = 20480, MA = 256, MB = 64, S32 = 80, S16 = 320;

    const float* pc        = (const float*)d_in[0];
    const float* node_a    = (const float*)d_in[1];
    const float* node_b    = (const float*)d_in[2];
    const float* img_g_a   = (const float*)d_in[3];   // (B,512,MA)
    const float* img_g_b   = (const float*)d_in[4];   // (B,512,MB)
    const float* img_s32   = (const float*)d_in[5];   // (B,512,S32)
    const float* img_s16   = (const float*)d_in[6];   // (B,256,S16)
    const float* nb_feat   = (const float*)d_in[7];   // (B,256,MB)
    const float* glob      = (const float*)d_in[8];   // (B,512,1)
    const float* na_feat   = (const float*)d_in[9];   // (B,64,MA)
    const int*   idx_pa    = (const int*)  d_in[10];  // (B,N,3)
    const float* nbatt_W0  = (const float*)d_in[11];
    const float* nbatt_b0  = (const float*)d_in[12];
    const float* nbatt_W1  = (const float*)d_in[13];
    const float* nbatt_b1  = (const float*)d_in[14];
    const float* nb_W0     = (const float*)d_in[15];
    const float* nb_b0     = (const float*)d_in[16];
    const float* nb_W1     = (const float*)d_in[17];
    const float* nb_b1     = (const float*)d_in[18];
    const float* nb_W2     = (const float*)d_in[19];
    const float* nb_b2     = (const float*)d_in[20];
    const float* naatt_W0  = (const float*)d_in[21];
    const float* naatt_b0  = (const float*)d_in[22];
    const float* naatt_W1  = (const float*)d_in[23];
    const float* naatt_b1  = (const float*)d_in[24];
    const float* na_W0     = (const float*)d_in[25];
    const float* na_b0     = (const float*)d_in[26];
    const float* na_W1     = (const float*)d_in[27];
    const float* na_b1     = (const float*)d_in[28];
    const float* na_W2     = (const float*)d_in[29];
    const float* na_b2     = (const float*)d_in[30];

    // Output layout: interp_pa (B,128,N) then interp_pb (B,512,N)
    float* out_pa = (float*)d_out;
    float* out_pb = out_pa + (size_t)B * 128 * N;

    // ---- workspace carve (floats; ints share the arena) -------------------
    float* ws = (float*)d_ws;
    size_t off = 0;
    auto alloc = [&](size_t n) { float* p = ws + off; off += n; return p; };

    float* cat_nbatt = alloc((size_t)B * 768 * MB);
    float* h_nbatt   = alloc((size_t)B * 256 * MB);
    float* attb      = alloc((size_t)B * S32 * MB);
    float* w_s32     = alloc((size_t)B * 512 * MB);
    float* cat_nb    = alloc((size_t)B * 1792 * MB);
    float* h_nb0     = alloc((size_t)B * 1024 * MB);
    float* h_nb1     = alloc((size_t)B * 512 * MB);
    float* up_b      = alloc((size_t)B * 512 * MB);
    float* cat_naatt = alloc((size_t)B * 576 * MA);
    float* h_naatt   = alloc((size_t)B * 256 * MA);
    float* atta      = alloc((size_t)B * S16 * MA);
    float* w_s16     = alloc((size_t)B * 256 * MA);
    float* interp_ab = alloc((size_t)B * 512 * MA);
    float* cat_na    = alloc((size_t)B * 832 * MA);
    float* h_na0     = alloc((size_t)B * 512 * MA);
    float* h_na1     = alloc((size_t)B * 128 * MA);
    float* up_a      = alloc((size_t)B * 128 * MA);
    int*   idx_pb    = (int*)alloc((size_t)B * N * 3);
    float* w_pb      = alloc((size_t)B * N * 3);
    int*   idx_ab    = (int*)alloc((size_t)B * MA * 3);
    float* w_ab      = alloc((size_t)B * MA * 3);
    float* w_pa      = alloc((size_t)B * N * 3);

    // ---- launch helpers ---------------------------------------------------
    auto copyBlk = [&](float* dst, int dstC, int dstOff, const float* src,
                       int C, int M, int srcM) {
        long total = (long)B * C * M;
        hipLaunchKernelGGL(copy_into, dim3((unsigned)((total + 255) / 256)),
                           dim3(256), 0, stream, dst, dstC, dstOff, src, C, M,
                           srcM, B);
    };
    auto gemm = [&](const float* A, long aStride, const float* X,
                    const float* bias, float* Y, int O, int I, int M, int relu) {
        hipLaunchKernelGGL(gemm_wmma_f32, dim3(M / 16, O / 16, B), dim3(32), 0,
                           stream, A, aStride, X, bias, Y, O, I, M, relu);
    };

    // ---- node_b attention branch -----------------------------------------
    copyBlk(cat_nbatt, 768, 0,   nb_feat, 256, MB, MB);
    copyBlk(cat_nbatt, 768, 256, img_g_b, 512, MB, MB);
    gemm(nbatt_W0, 0, cat_nbatt, nbatt_b0, h_nbatt, 256, 768, MB, 1);
    gemm(nbatt_W1, 0, h_nbatt,   nbatt_b1, attb,    S32, 256, MB, 0);
    hipLaunchKernelGGL(softmax_ch, dim3((B * MB + 255) / 256), dim3(256), 0,
                       stream, attb, S32, MB, B);
    // w_s32 = img_s32 (batched A) * att_b
    gemm(img_s32, (long)512 * S32, attb, nullptr, w_s32, 512, S32, MB, 0);

    // ---- up_b MLP ---------------------------------------------------------
    copyBlk(cat_nb, 1792, 0,    nb_feat, 256, MB, MB);
    copyBlk(cat_nb, 1792, 256,  glob,    512, MB, 1);   // broadcast along M
    copyBlk(cat_nb, 1792, 768,  w_s32,   512, MB, MB);
    copyBlk(cat_nb, 1792, 1280, img_g_b, 512, MB, MB);
    gemm(nb_W0, 0, cat_nb, nb_b0, h_nb0, 1024, 1792, MB, 1);
    gemm(nb_W1, 0, h_nb0,  nb_b1, h_nb1, 512,  1024, MB, 1);
    gemm(nb_W2, 0, h_nb1,  nb_b2, up_b,  512,  512,  MB, 0);

    // ---- pc -> node_b interpolation (2nd output) --------------------------
    hipLaunchKernelGGL(topk3_weights, dim3((B * N + 255) / 256), dim3(256), 0,
                       stream, pc, node_b, N, MB, idx_pb, w_pb, B);
    hipLaunchKernelGGL(interp3, dim3((N + 255) / 256, 512, B), dim3(256), 0,
                       stream, out_pb, up_b, idx_pb, w_pb, 512, N, MB, B);

    // ---- node_a attention branch -----------------------------------------
    copyBlk(cat_naatt, 576, 0,  na_feat, 64,  MA, MA);
    copyBlk(cat_naatt, 576, 64, img_g_a, 512, MA, MA);
    gemm(naatt_W0, 0, cat_naatt, naatt_b0, h_naatt, 256, 576, MA, 1);
    gemm(naatt_W1, 0, h_naatt,   naatt_b1, atta,    S16, 256, MA, 0);
    hipLaunchKernelGGL(softmax_ch, dim3((B * MA + 255) / 256), dim3(256), 0,
                       stream, atta, S16, MA, B);
    gemm(img_s16, (long)256 * S16, atta, nullptr, w_s16, 256, S16, MA, 0);

    // ---- node_a -> node_b interpolation -----------------------------------
    hipLaunchKernelGGL(topk3_weights, dim3((B * MA + 255) / 256), dim3(256), 0,
                       stream, node_a, node_b, MA, MB, idx_ab, w_ab, B);
    hipLaunchKernelGGL(interp3, dim3((MA + 255) / 256, 512, B), dim3(256), 0,
                       stream, interp_ab, up_b, idx_ab, w_ab, 512, MA, MB, B);

    // ---- up_a MLP ---------------------------------------------------------
    copyBlk(cat_na, 832, 0,   na_feat,   64,  MA, MA);
    copyBlk(cat_na, 832, 64,  interp_ab, 512, MA, MA);
    copyBlk(cat_na, 832, 576, w_s16,     256, MA, MA);
    gemm(na_W0, 0, cat_na, na_b0, h_na0, 512, 832, MA, 1);
    gemm(na_W1, 0, h_na0,  na_b1, h_na1, 128, 512, MA, 1);
    gemm(na_W2, 0, h_na1,  na_b2, up_a,  128, 128, MA, 0);

    // ---- pc -> node_a interpolation (1st output, indices precomputed) -----
    hipLaunchKernelGGL(weights_from_idx, dim3((B * N + 255) / 256), dim3(256),
                       0, stream, pc, node_a, idx_pa, N, MA, w_pa, B);
    hipLaunchKernelGGL(interp3, dim3((N + 255) / 256, 128, B), dim3(256), 0,
                       stream, out_pa, up_a, idx_pa, w_pa, 128, N, MA, B);
}